// Focus_72894184948282
// MI455X (gfx1250) — compile-verified
//
#include <hip/hip_runtime.h>
#include <math.h>

typedef __attribute__((ext_vector_type(2))) float v2f;
typedef __attribute__((ext_vector_type(4))) float v4f;
typedef __attribute__((ext_vector_type(8))) float v8f;

#define IMG_START 35
#define NUM_FRAMES 8
#define PATCH_H 24
#define PATCH_W 24
#define IMG_LEN (NUM_FRAMES * PATCH_H * PATCH_W)   // 4608
#define SEQ_LEN (IMG_START + IMG_LEN + 100)        // 4743
#define HIDDEN 4096
#define VEC 32
#define VNUM (HIDDEN / VEC)                        // 128
#define NTILES (IMG_LEN / 16)                      // 288

// ---------------------------------------------------------------------------
// Pass 1: per-chunk squared norms into workspace. chunk idx = t*128 + v, and
// the chunk's 32 floats are contiguous at img + idx*32.
// ---------------------------------------------------------------------------
__global__ __launch_bounds__(256) void norms_kernel(const float* __restrict__ img,
                                                    float* __restrict__ n2) {
    int idx = blockIdx.x * 256 + threadIdx.x;      // [0, 4608*128)
    const float* p = img + (size_t)idx * VEC;
    float s = 0.f;
#pragma unroll
    for (int c = 0; c < 8; ++c) {
        v4f q = *(const v4f*)(p + c * 4);
        s += q.x * q.x + q.y * q.y + q.z * q.z + q.w * q.w;
    }
    n2[idx] = s;
}

// ---------------------------------------------------------------------------
// Copy the non-image prefix (35 rows) and suffix (100 rows) as float4.
// 35*1024 + 100*1024 = 138240 float4 = 540 blocks * 256 threads.
// ---------------------------------------------------------------------------
__global__ __launch_bounds__(256) void copy_edges_kernel(const float* __restrict__ in,
                                                         float* __restrict__ out) {
    int idx = blockIdx.x * 256 + threadIdx.x;
    const int npre = IMG_START * (HIDDEN / 4);     // 35840 float4
    size_t fo;
    if (idx < npre) {
        fo = (size_t)idx * 4;
    } else {
        fo = (size_t)(IMG_START + IMG_LEN) * HIDDEN + (size_t)(idx - npre) * 4;
    }
    *(v4f*)(out + fo) = *(const v4f*)(in + fo);
}

// ---------------------------------------------------------------------------
// Pass 2: one wave per (16-token tile, chunk v). Computes diag(A * B_off^T)
// with V_WMMA_F32_16X16X4_F32 over 8 K-slices per offset.
//
// Fragment layout (f32, K=4): lane L<16 holds row (t0 + L) floats {4c,4c+1};
// lane L+16 holds floats {4c+2,4c+3}. Identical selection pattern for A
// (lane=M) and B (lane=N), so one loader serves both operands.
//
// Diagonal of the 16x16 f32 D tile: token i<8 -> VGPR i, lane i;
// token i>=8 -> VGPR i-8, lane i+16. So valid lanes are 0-7 and 24-31;
// each extracts acc[lane&7].
// ---------------------------------------------------------------------------
__global__ __launch_bounds__(256) void focus_wmma_kernel(const float* __restrict__ img_in,
                                                         float* __restrict__ img_out,
                                                         const float* __restrict__ n2) {
    const int lane = threadIdx.x & 31;
    const int wave = (blockIdx.x * 256 + threadIdx.x) >> 5;   // [0, 288*128)
    const int tile = wave >> 7;                               // t-tile [0,288)
    const int v    = wave & 127;                              // chunk  [0,128)
    const int t0   = tile * 16;

    const int m  = lane & 15;            // row-in-tile this lane loads/stores
    const int hi = (lane >> 4) & 1;      // 0: K pair {4c,4c+1}; 1: {4c+2,4c+3}

    // ---- load self fragment (A operand + store source) ----
    const float* sp = img_in + (size_t)(t0 + m) * HIDDEN + v * VEC + hi * 2;
    v2f a[8];
#pragma unroll
    for (int c = 0; c < 8; ++c) a[c] = *(const v2f*)(sp + c * 4);

    // token index this lane is responsible for in the diag-extraction phase:
    // lanes 0-7 -> tokens 0-7, lanes 24-31 -> tokens 8-15 (others are junk but
    // map to a valid in-tile token so loads stay in bounds).
    const int i_l = (lane & 7) + ((lane & 16) >> 1);
    const int ti  = t0 + i_l;
    const float n2a = n2[ti * VNUM + v];

    unsigned my_masked = 0u;
    const int offs[7] = {1, 24, 25, 576, 577, 600, 601};

#pragma unroll
    for (int oi = 0; oi < 7; ++oi) {
        const int off = offs[oi];

        // ---- load neighbor fragment (B operand), clamp rows below 0 ----
        int rn = t0 + m - off;
        int rc = rn < 0 ? 0 : rn;
        const float* nbp = img_in + (size_t)rc * HIDDEN + v * VEC + hi * 2;
        v2f b[8];
#pragma unroll
        for (int c = 0; c < 8; ++c) b[c] = *(const v2f*)(nbp + c * 4);

        // ---- 8 chained WMMAs: D = sum_c A_c * B_c ----
        v8f acc = {};
#pragma unroll
        for (int c = 0; c < 8; ++c)
            acc = __builtin_amdgcn_wmma_f32_16x16x4_f32(
                false, a[c], false, b[c], (short)0, acc, false, false);

        // ---- extract this lane's diagonal element ----
        const int s = lane & 7;
        float dot = acc[0];
#pragma unroll
        for (int j = 1; j < 8; ++j) dot = (s == j) ? acc[j] : dot;

        // ---- cosine threshold test (shift-in-zero rows give sim 0) ----
        int tn = ti - off;
        bool valid = tn >= 0;
        float n2b = n2[(valid ? tn : 0) * VNUM + v];
        float th = 0.9f * sqrtf(n2a * n2b);
        my_masked |= (valid && (dot > th)) ? 1u : 0u;
    }

    // broadcast the 16 per-token decisions to the whole wave
    unsigned bal = __builtin_amdgcn_ballot_w32(my_masked != 0u);
    int bit = (m < 8) ? m : (m + 16);
    bool zero_me = (bal >> bit) & 1u;

    // ---- store: original fragment or zeros ----
    float* op = img_out + (size_t)(t0 + m) * HIDDEN + v * VEC + hi * 2;
#pragma unroll
    for (int c = 0; c < 8; ++c) {
        v2f val = a[c];
        if (zero_me) { val.x = 0.f; val.y = 0.f; }
        *(v2f*)(op + c * 4) = val;
    }
}

extern "C" void kernel_launch(void* const* d_in, const int* in_sizes, int n_in,
                              void* d_out, int out_size, void* d_ws, size_t ws_size,
                              hipStream_t stream) {
    (void)in_sizes; (void)n_in; (void)out_size; (void)ws_size;
    const float* x  = (const float*)d_in[0];
    float* out      = (float*)d_out;
    float* n2       = (float*)d_ws;                 // 4608*128 floats = 2.36 MB

    const float* img_in  = x   + (size_t)IMG_START * HIDDEN;
    float*       img_out = out + (size_t)IMG_START * HIDDEN;

    // Pass 1: norms (4608*128 = 589824 threads)
    norms_kernel<<<(IMG_LEN * VNUM) / 256, 256, 0, stream>>>(img_in, n2);

    // Edge copy: 138240 float4 = 540 blocks
    copy_edges_kernel<<<540, 256, 0, stream>>>(x, out);

    // Pass 2: 288 tiles * 128 chunks = 36864 waves = 4608 blocks of 8 waves
    focus_wmma_kernel<<<(NTILES * VNUM) / 8, 256, 0, stream>>>(img_in, img_out, n2);
}